// GraphSAGE_87110526698154
// MI455X (gfx1250) — compile-verified
//
#include <hip/hip_runtime.h>
#include <hip/hip_bf16.h>

typedef float v2f __attribute__((ext_vector_type(2)));
typedef float v8f __attribute__((ext_vector_type(8)));

#define N_NODES 50000
#define N_EDGES 800000
#define F_IN    128
#define H_DIM   256
#define C_DIM   64

// ---------------------------------------------------------------- zero fill
__global__ void k_zero(float* __restrict__ p, size_t n) {
    size_t i = (size_t)blockIdx.x * blockDim.x + threadIdx.x;
    size_t stride = (size_t)gridDim.x * blockDim.x;
    for (; i < n; i += stride) p[i] = 0.0f;
}

// ---------------------------------------------------------------- in-degree
__global__ void k_degree(const long long* __restrict__ dst,
                         float* __restrict__ deg, int E) {
    int e = blockIdx.x * blockDim.x + threadIdx.x;
    if (e < E) atomicAdd(&deg[(int)dst[e]], 1.0f);
}

// ------------------------------------------------ scatter-add: agg[dst] += feat[src]
// One thread handles one float4 chunk of one edge; consecutive threads walk the
// feature dim of the same edge -> coalesced float4 gathers, L2-resident atomics.
template <int FD>
__global__ void k_scatter(const long long* __restrict__ src,
                          const long long* __restrict__ dst,
                          const float* __restrict__ feat,
                          float* __restrict__ agg, int E) {
    constexpr int CHUNKS = FD / 4;
    long long t = (long long)blockIdx.x * blockDim.x + threadIdx.x;
    if (t >= (long long)E * CHUNKS) return;
    int e = (int)(t >> (__builtin_ctz(CHUNKS)));      // t / CHUNKS (power of 2)
    int c = (int)(t & (CHUNKS - 1));                  // t % CHUNKS
    int s = (int)src[e];
    int d = (int)dst[e];
    const float4 v = reinterpret_cast<const float4*>(feat + (size_t)s * FD)[c];
    float* o = agg + (size_t)d * FD + (size_t)c * 4;
    atomicAdd(o + 0, v.x);
    atomicAdd(o + 1, v.y);
    atomicAdd(o + 2, v.z);
    atomicAdd(o + 3, v.w);
}

// ------------------------------------------------ fused SAGE layer via f32 WMMA
// out[16x16 tile] = act( (agg/max(deg,1)) @ Wl  +  self @ Wr  + bias )
// One wave (32 threads) per 16x16 output tile. K-loop of V_WMMA_F32_16X16X4_F32.
//
// A-fragment (16x4 f32): lane<16 holds A[M=lane][K=kb..kb+1] with kb=0;
//                        lane>=16 holds A[M=lane-16][K=2..3].
// B-fragment (4x16 f32): lane holds B[K=kb..kb+1][N=lane%16], kb=(lane/16)*2.
// C/D (16x16 f32, 8 VGPRs): vgpr j, lane<16 -> (M=j, N=lane);
//                           lane>=16 -> (M=j+8, N=lane-16).
template <int K, int HD, bool RELU>
__global__ void sage_layer_wmma(const float* __restrict__ agg,
                                const float* __restrict__ deg,
                                const float* __restrict__ self_feat,
                                const float* __restrict__ Wl,   // [K, HD]
                                const float* __restrict__ Wr,   // [K, HD]
                                const float* __restrict__ bias, // [HD]
                                float* __restrict__ out) {      // [N, HD]
    const int lane = threadIdx.x & 31;
    const int half = lane >> 4;              // 0 or 1
    const int m    = blockIdx.y * 16 + (lane & 15);
    const int n    = blockIdx.x * 16 + (lane & 15);
    const int koff = half * 2;               // lane-group K offset within step

    const float scale = 1.0f / fmaxf(deg[m], 1.0f);
    const float* aggRow  = agg       + (size_t)m * K;
    const float* selfRow = self_feat + (size_t)m * K;

    v8f acc = {};
    for (int k0 = 0; k0 < K; k0 += 4) {
        const float2 av = *reinterpret_cast<const float2*>(aggRow  + k0 + koff);
        const float2 sv = *reinterpret_cast<const float2*>(selfRow + k0 + koff);
        v2f a1; a1.x = av.x * scale; a1.y = av.y * scale;
        v2f a2; a2.x = sv.x;         a2.y = sv.y;
        v2f b1; b1.x = Wl[(size_t)(k0 + koff) * HD + n];
                b1.y = Wl[(size_t)(k0 + koff + 1) * HD + n];
        v2f b2; b2.x = Wr[(size_t)(k0 + koff) * HD + n];
                b2.y = Wr[(size_t)(k0 + koff + 1) * HD + n];
        acc = __builtin_amdgcn_wmma_f32_16x16x4_f32(
                  false, a1, false, b1, (short)0, acc, false, false);
        acc = __builtin_amdgcn_wmma_f32_16x16x4_f32(
                  false, a2, false, b2, (short)0, acc, false, false);
    }

    const float bn    = bias[n];
    const int   mbase = blockIdx.y * 16 + half * 8;
#pragma unroll
    for (int j = 0; j < 8; ++j) {
        float v = acc[j] + bn;
        if (RELU) v = fmaxf(v, 0.0f);
        out[(size_t)(mbase + j) * HD + n] = v;
    }
}

// ---------------------------------------------------------------- log_softmax (64 cols)
// One wave per row; lane handles columns lane and lane+32. In-place.
__global__ void k_log_softmax64(float* __restrict__ data, int nrows) {
    const int wavesPerBlock = blockDim.x >> 5;
    const int row  = blockIdx.x * wavesPerBlock + (threadIdx.x >> 5);
    const int lane = threadIdx.x & 31;
    if (row >= nrows) return;
    float* p = data + (size_t)row * C_DIM;
    float v0 = p[lane];
    float v1 = p[lane + 32];
    float mx = fmaxf(v0, v1);
#pragma unroll
    for (int o = 16; o > 0; o >>= 1) mx = fmaxf(mx, __shfl_xor(mx, o, 32));
    float s = expf(v0 - mx) + expf(v1 - mx);
#pragma unroll
    for (int o = 16; o > 0; o >>= 1) s += __shfl_xor(s, o, 32);
    const float lse = mx + logf(s);
    p[lane]      = v0 - lse;
    p[lane + 32] = v1 - lse;
}

// ---------------------------------------------------------------- launch
extern "C" void kernel_launch(void* const* d_in, const int* in_sizes, int n_in,
                              void* d_out, int out_size, void* d_ws, size_t ws_size,
                              hipStream_t stream) {
    const float*     x   = (const float*)d_in[0];
    const long long* ei  = (const long long*)d_in[1];   // int64 per reference
    const float*     W1l = (const float*)d_in[2];
    const float*     b1  = (const float*)d_in[3];
    const float*     W1r = (const float*)d_in[4];
    const float*     W2l = (const float*)d_in[5];
    const float*     b2  = (const float*)d_in[6];
    const float*     W2r = (const float*)d_in[7];
    float*           out = (float*)d_out;

    const long long* src = ei;            // edge_index[0]
    const long long* dst = ei + N_EDGES;  // edge_index[1]

    // workspace layout (floats): deg[N] | agg1[N*F] | agg2[N*H] | h[N*H]
    float* ws   = (float*)d_ws;
    float* deg  = ws;
    float* agg1 = deg  + N_NODES;
    float* agg2 = agg1 + (size_t)N_NODES * F_IN;
    float* h    = agg2 + (size_t)N_NODES * H_DIM;

    // 1) zero deg + agg1 + agg2 (contiguous region)
    const size_t zn = (size_t)N_NODES * (1 + F_IN + H_DIM);
    k_zero<<<4096, 256, 0, stream>>>(ws, zn);

    // 2) in-degree
    k_degree<<<(N_EDGES + 255) / 256, 256, 0, stream>>>(dst, deg, N_EDGES);

    // 3) agg1 += x[src]
    {
        long long total = (long long)N_EDGES * (F_IN / 4);
        k_scatter<F_IN><<<(unsigned)((total + 255) / 256), 256, 0, stream>>>(
            src, dst, x, agg1, N_EDGES);
    }

    // 4) h = relu(mean1 @ W1_l + x @ W1_r + b1)   [N, 256]
    sage_layer_wmma<F_IN, H_DIM, true>
        <<<dim3(H_DIM / 16, N_NODES / 16), 32, 0, stream>>>(
            agg1, deg, x, W1l, W1r, b1, h);

    // 5) agg2 += h[src]
    {
        long long total = (long long)N_EDGES * (H_DIM / 4);
        k_scatter<H_DIM><<<(unsigned)((total + 255) / 256), 256, 0, stream>>>(
            src, dst, h, agg2, N_EDGES);
    }

    // 6) logits = mean2 @ W2_l + h @ W2_r + b2 -> d_out   [N, 64]
    sage_layer_wmma<H_DIM, C_DIM, false>
        <<<dim3(C_DIM / 16, N_NODES / 16), 32, 0, stream>>>(
            agg2, deg, h, W2l, W2r, b2, out);

    // 7) in-place log_softmax over 64 columns
    k_log_softmax64<<<(N_NODES + 7) / 8, 256, 0, stream>>>(out, N_NODES);
}